// ChenAllocator_26268019982506
// MI455X (gfx1250) — compile-verified
//
#include <hip/hip_runtime.h>
#include <math.h>

typedef float v2f __attribute__((ext_vector_type(2)));
typedef float v8f __attribute__((ext_vector_type(8)));

#define ITERS 200

// denom = (1<<bits)-1 for bits {2,3,4,5,6,7,8,16}
// C[i][j] = 0.5*trH*(2*wmax/denom)^2/12 = trH*wmax^2 / (6*denom^2)
__constant__ float c_inv6d2[8] = {
    1.0f / (6.0f * 3.0f * 3.0f),
    1.0f / (6.0f * 7.0f * 7.0f),
    1.0f / (6.0f * 15.0f * 15.0f),
    1.0f / (6.0f * 31.0f * 31.0f),
    1.0f / (6.0f * 63.0f * 63.0f),
    1.0f / (6.0f * 127.0f * 127.0f),
    1.0f / (6.0f * 255.0f * 255.0f),
    1.0f / (6.0f * 65535.0f * 65535.0f)
};

__global__ __launch_bounds__(32, 1)
void sinkhorn_wave_kernel(const float* __restrict__ theta,  // [64,8]
                          const float* __restrict__ phi,    // [8]
                          const float* __restrict__ trH,    // [64]
                          const float* __restrict__ wmax,   // [64]
                          const float* __restrict__ a,      // [64]
                          float* __restrict__ out)          // [64,8]
{
    const int lane = threadIdx.x;          // 0..31, one full wave32
    const int r0 = lane;                   // row lane
    const int r1 = lane + 32;              // row lane+32

    // ---- Build K = (theta - C) / EPS, EPS = 0.02 -> *50 ----
    const float tw0 = trH[r0] * wmax[r0] * wmax[r0];
    const float tw1 = trH[r1] * wmax[r1] * wmax[r1];

    float K0[8], K1[8];
#pragma unroll
    for (int j = 0; j < 8; ++j) {
        const float inv = c_inv6d2[j];
        K0[j] = (theta[r0 * 8 + j] - tw0 * inv) * 50.0f;
        K1[j] = (theta[r1 * 8 + j] - tw1 * inv) * 50.0f;
    }

    const float log_a0 = __logf(a[r0] + 1e-40f);
    const float log_a1 = __logf(a[r1] + 1e-40f);

    // ---- log_b = log(softmax(phi) + 1e-40), computed redundantly per lane ----
    float ph[8];
#pragma unroll
    for (int j = 0; j < 8; ++j) ph[j] = phi[j];
    float m = ph[0];
#pragma unroll
    for (int j = 1; j < 8; ++j) m = fmaxf(m, ph[j]);
    float eb[8], se = 0.0f;
#pragma unroll
    for (int j = 0; j < 8; ++j) { eb[j] = __expf(ph[j] - m); se += eb[j]; }
    const float inv_se = 1.0f / se;
    float log_b[8];
#pragma unroll
    for (int j = 0; j < 8; ++j) log_b[j] = __logf(eb[j] * inv_se + 1e-40f);

    // ---- Sinkhorn iterations (latency chain; all in VGPRs, no barriers) ----
    float g[8];
#pragma unroll
    for (int j = 0; j < 8; ++j) g[j] = 0.0f;
    float f0 = 0.0f, f1 = 0.0f;

    for (int it = 0; it < ITERS; ++it) {
        // f update: lane-local logsumexp over 8 columns (each of 2 rows)
        float s0 = 0.0f, s1 = 0.0f;
#pragma unroll
        for (int j = 0; j < 8; ++j) {
            s0 += __expf(K0[j] + g[j]);
            s1 += __expf(K1[j] + g[j]);
        }
        f0 = log_a0 - __logf(s0);
        f1 = log_a1 - __logf(s1);

        // g update: column sums over 64 rows = 2 local rows + 5-step wave32 xor tree
        float p[8];
#pragma unroll
        for (int j = 0; j < 8; ++j)
            p[j] = __expf(K0[j] + f0) + __expf(K1[j] + f1);
#pragma unroll
        for (int mask = 1; mask < 32; mask <<= 1) {
#pragma unroll
            for (int j = 0; j < 8; ++j)
                p[j] += __shfl_xor(p[j], mask, 32);
        }
#pragma unroll
        for (int j = 0; j < 8; ++j)
            g[j] = log_b[j] - __logf(p[j]);
    }

    // ---- P = exp(K + f + g) ----
    float P0[8], P1[8];
#pragma unroll
    for (int j = 0; j < 8; ++j) {
        P0[j] = __expf(K0[j] + f0 + g[j]);
        P1[j] = __expf(K1[j] + f1 + g[j]);
    }

    // ---- Global sum of all 512 P values via chained V_WMMA_F32_16X16X4_F32 ----
    // With A = ones(16x4), D[m][n] = sum_k B[k][n]; B's K dim maps to
    // (vgpr pair) x (lane, lane+16). Eight accumulating WMMAs fold all 16
    // per-lane P registers plus both lane halves into D[*][n] = colsum(n),
    // replicated across all D rows -> every lane's acc[0] holds colsum(lane%16).
    v2f ones; ones[0] = 1.0f; ones[1] = 1.0f;
    v8f acc = {};
#pragma unroll
    for (int t = 0; t < 4; ++t) {
        v2f b; b[0] = P0[2 * t]; b[1] = P0[2 * t + 1];
        acc = __builtin_amdgcn_wmma_f32_16x16x4_f32(
            false, ones, false, b, (short)0, acc, false, false);
    }
#pragma unroll
    for (int t = 0; t < 4; ++t) {
        v2f b; b[0] = P1[2 * t]; b[1] = P1[2 * t + 1];
        acc = __builtin_amdgcn_wmma_f32_16x16x4_f32(
            false, ones, false, b, (short)0, acc, false, false);
    }
    float tot = acc[0];                     // colsum(lane%16), identical in both halves
    tot += __shfl_xor(tot, 1, 32);
    tot += __shfl_xor(tot, 2, 32);
    tot += __shfl_xor(tot, 4, 32);
    tot += __shfl_xor(tot, 8, 32);          // now full 512-element sum in every lane

    const float inv_tot = 1.0f / (tot + 1e-40f);
#pragma unroll
    for (int j = 0; j < 8; ++j) {
        out[r0 * 8 + j] = P0[j] * inv_tot;
        out[r1 * 8 + j] = P1[j] * inv_tot;
    }
}

extern "C" void kernel_launch(void* const* d_in, const int* in_sizes, int n_in,
                              void* d_out, int out_size, void* d_ws, size_t ws_size,
                              hipStream_t stream) {
    (void)in_sizes; (void)n_in; (void)d_ws; (void)ws_size; (void)out_size;
    const float* theta = (const float*)d_in[0];
    const float* phi   = (const float*)d_in[1];
    const float* trH   = (const float*)d_in[2];
    const float* wmax  = (const float*)d_in[3];
    const float* a     = (const float*)d_in[4];
    float* out = (float*)d_out;

    sinkhorn_wave_kernel<<<1, 32, 0, stream>>>(theta, phi, trH, wmax, a, out);
}